// SSM_83846351552556
// MI455X (gfx1250) — compile-verified
//
#include <hip/hip_runtime.h>

typedef __attribute__((ext_vector_type(2))) float v2f;
typedef __attribute__((ext_vector_type(8))) float v8f;

#define B_    2
#define D_    128
#define N_    64
#define T_    4096
#define BD    256        // B_*D_
#define TP1   4097       // T_+1
#define CHUNK 16
#define NCH   256        // T_/CHUNK

// Workspace layout (floats):
//   Mt  : [N_][NCH][16][16]  = 4,194,304 f32 (16 MB)   Mt[n][c][j][i] = M[i][j]
//   Pfx : [N_][NCH][16]      =   262,144 f32 ( 1 MB)   Pfx[n][c][i] = prod_{u<=i} a
//   E   : [N_][NCH][BD]      = 4,194,304 f32 (16 MB)   chunk entry states
#define MT_ELEMS  (N_ * NCH * 16 * 16)
#define PFX_ELEMS (N_ * NCH * 16)

// ---------------------------------------------------------------------------
// Kernel 0: build per-(n,c) transposed decay matrix Mt and prefix products.
//   M[i][j] = prod_{u=j+1}^{i} a[u]  (i>j), 1 (i==j), 0 (i<j);  a = A_bar chunk
// One thread per (n,c); rows stored with float4 (64B-aligned).
// ---------------------------------------------------------------------------
__global__ __launch_bounds__(256) void ssm_build_mt_kernel(
    const float* __restrict__ Ab, float* __restrict__ Mt,
    float* __restrict__ Pfx)
{
    const int n = blockIdx.x;       // 0..63
    const int c = threadIdx.x;      // 0..255

    float a[16];
    #pragma unroll
    for (int u = 0; u < 16; ++u) a[u] = Ab[n * T_ + c * CHUNK + u];

    float* mrow = Mt + (size_t)((n * NCH + c) * 16) * 16;
    #pragma unroll
    for (int j = 0; j < 16; ++j) {
        float row[16];
        float r = 1.0f;
        #pragma unroll
        for (int i = 0; i < 16; ++i) {
            if (i < j)       row[i] = 0.0f;
            else if (i == j) row[i] = 1.0f;
            else             { r *= a[i]; row[i] = r; }
        }
        float4* p = (float4*)(mrow + j * 16);
        p[0] = make_float4(row[0],  row[1],  row[2],  row[3]);
        p[1] = make_float4(row[4],  row[5],  row[6],  row[7]);
        p[2] = make_float4(row[8],  row[9],  row[10], row[11]);
        p[3] = make_float4(row[12], row[13], row[14], row[15]);
    }

    float* pp = Pfx + (size_t)(n * NCH + c) * 16;
    float p = 1.0f;
    #pragma unroll
    for (int i = 0; i < 16; ++i) { p *= a[i]; pp[i] = p; }
}

// ---------------------------------------------------------------------------
// Kernel 1: per-(n,bd) sequential scan over chunk summaries.
//   E[n,c,bd] = s_{16c};  E_{c+1} = p16*E_c + B_bar[n]*(row15(M) . ht-chunk)
// ---------------------------------------------------------------------------
__global__ __launch_bounds__(256) void ssm_carry_kernel(
    const float* __restrict__ ht, const float* __restrict__ Ab,
    const float* __restrict__ Bb, float* __restrict__ Ews)
{
    const int n  = blockIdx.x;      // 0..63
    const int bd = threadIdx.x;     // 0..255
    const float Bn = Bb[n];

    float E = Bn * ht[bd * TP1];    // s_0 = b_0
    Ews[(n * NCH + 0) * BD + bd] = E;

    for (int c = 0; c < NCH - 1; ++c) {
        const float* Ap = Ab + n * T_ + c * CHUNK;      // uniform per block
        const float* hp = ht + bd * TP1 + c * CHUNK + 1;
        __builtin_prefetch((const void*)(hp + CHUNK), 0, 0);  // global_prefetch_b8

        float a[16], h[16];
        #pragma unroll
        for (int u = 0; u < 16; ++u) a[u] = Ap[u];
        #pragma unroll
        for (int u = 0; u < 16; ++u) h[u] = hp[u];

        float r = 1.0f;
        float dot = h[15];
        #pragma unroll
        for (int j = 14; j >= 0; --j) { r *= a[j + 1]; dot += r * h[j]; }
        const float p16 = r * a[0];

        E = p16 * E + Bn * dot;
        Ews[(n * NCH + c + 1) * BD + bd] = E;
    }
}

// ---------------------------------------------------------------------------
// Kernel 2: intra-chunk scan as FP32 WMMA GEMM + epilogue + y reduction.
//   D[bd', i] = sum_j Ht[bd', j] * M[i][j]   (A = Ht chunk rows, B = M^T)
//   h = B_bar[n]*D + Pfx[i]*E[n,c,bd];  y = sum_n C*h (LDS cross-wave reduce)
// Grid: (NCH, 16 bd-tiles) x 256 thr (8 waves; wave w -> n = 8w..8w+7)
// ---------------------------------------------------------------------------
__global__ __launch_bounds__(256) void ssm_chunk_wmma_kernel(
    const float* __restrict__ ht, const float* __restrict__ Bb,
    const float* __restrict__ Cc, const float* __restrict__ Mt,
    const float* __restrict__ Pfx, const float* __restrict__ Ews,
    float* __restrict__ hOut, float* __restrict__ yOut)
{
    __shared__ float yl[8][16][16];

    const int c    = blockIdx.x;          // chunk
    const int bd0  = blockIdx.y * 16;     // bd tile base
    const int bIdx = bd0 >> 7;            // batch index (tiles don't straddle B)
    const int tid  = threadIdx.x;
    const int wave = tid >> 5;            // wave32
    const int lane = tid & 31;
    const int half = lane >> 4;
    const int l16  = lane & 15;
    const int t0   = c * CHUNK;

    // A-fragments: A[M=bd'=l16][K], VGPR v holds K = 4s + v + 2*half.
    // Direct lane-offset loads (no register array, no selects).
    const float* hp = ht + (size_t)(bd0 + l16) * TP1 + t0 + 1 + 2 * half;
    v2f fragA[4];
    #pragma unroll
    for (int s = 0; s < 4; ++s) {
        fragA[s].x = hp[4 * s + 0];
        fragA[s].y = hp[4 * s + 1];
    }

    float yacc[8];
    #pragma unroll
    for (int v = 0; v < 8; ++v) yacc[v] = 0.0f;

    for (int q = 0; q < 8; ++q) {
        const int n = wave * 8 + q;

        // B-fragments from precomputed Mt: B[K=j][N=i=l16], row j = 4s+v+2*half
        const float* mb = Mt + (size_t)((n * NCH + c) * 16) * 16 + 2 * half * 16 + l16;
        v8f acc = {0.f, 0.f, 0.f, 0.f, 0.f, 0.f, 0.f, 0.f};
        #pragma unroll
        for (int s = 0; s < 4; ++s) {
            v2f fb;
            fb.x = mb[(4 * s + 0) * 16];
            fb.y = mb[(4 * s + 1) * 16];
            acc = __builtin_amdgcn_wmma_f32_16x16x4_f32(
                /*neg_a=*/false, fragA[s], /*neg_b=*/false, fb,
                /*c_mod=*/(short)0, acc, /*reuse_a=*/false, /*reuse_b=*/false);
        }

        const float pf = Pfx[(size_t)(n * NCH + c) * 16 + l16];
        const float Bn = Bb[n];
        const float Cv = Cc[(size_t)(bIdx * N_ + n) * T_ + t0 + l16];
        const float* Ep = Ews + (size_t)(n * NCH + c) * BD + bd0;

        // D layout: VGPR v -> row bd' = v + 8*half, col = l16 (t within chunk)
        #pragma unroll
        for (int v = 0; v < 8; ++v) {
            const int bdp = v + 8 * half;
            const float Ev = Ep[bdp];
            const float hval = Bn * acc[v] + pf * Ev;
            hOut[((size_t)(bd0 + bdp) * N_ + n) * T_ + t0 + l16] = hval;
            yacc[v] += Cv * hval;
        }
    }

    // y reduction across the 8 waves (each wave covered a disjoint n-set)
    #pragma unroll
    for (int v = 0; v < 8; ++v) yl[wave][v + 8 * half][l16] = yacc[v];
    __syncthreads();
    {
        const int bdp = tid >> 4;
        const int i   = tid & 15;
        float s = 0.0f;
        #pragma unroll
        for (int w = 0; w < 8; ++w) s += yl[w][bdp][i];
        yOut[(size_t)(bd0 + bdp) * T_ + t0 + i] = s;
    }
}

// ---------------------------------------------------------------------------
extern "C" void kernel_launch(void* const* d_in, const int* in_sizes, int n_in,
                              void* d_out, int out_size, void* d_ws, size_t ws_size,
                              hipStream_t stream) {
    const float* ht = (const float*)d_in[0];   // (B,D,1,T+1)
    const float* Ab = (const float*)d_in[1];   // (N,T)
    const float* Bb = (const float*)d_in[2];   // (N,1)
    const float* Cc = (const float*)d_in[3];   // (B,N,T)

    float* hOut = (float*)d_out;                             // (B,D,N,T)
    float* yOut = hOut + (size_t)B_ * D_ * N_ * T_;          // (B,D,T)

    float* Mt  = (float*)d_ws;                               // 16 MB
    float* Pfx = Mt + MT_ELEMS;                              //  1 MB
    float* Ews = Pfx + PFX_ELEMS;                            // 16 MB

    ssm_build_mt_kernel<<<dim3(N_), dim3(NCH), 0, stream>>>(Ab, Mt, Pfx);
    ssm_carry_kernel<<<dim3(N_), dim3(BD), 0, stream>>>(ht, Ab, Bb, Ews);
    ssm_chunk_wmma_kernel<<<dim3(NCH, BD / 16), dim3(256), 0, stream>>>(
        ht, Bb, Cc, Mt, Pfx, Ews, hOut, yOut);
}